// SWAttention_AV_60730837565791
// MI455X (gfx1250) — compile-verified
//
#include <hip/hip_runtime.h>

// Problem constants from the reference.
#define B_    2
#define NH_   8
#define N_    4096
#define D_    64
#define K2_   49
#define KPAD  64          // k2 padded to multiple of 16 for WMMA column tiles
#define TILE_M 16         // pixels per workgroup

typedef __attribute__((ext_vector_type(2))) float v2f;
typedef __attribute__((ext_vector_type(8))) float v8f;

__global__ __launch_bounds__(256)
void swattn_av_fused(const float* __restrict__ q,          // [B,NH,N,D]
                     const float* __restrict__ attn_local, // [B,NH,N,K2]
                     const float* __restrict__ v_local,    // [B,NH,N,D,K2]
                     const float* __restrict__ tokens,     // [NH,D,K2]
                     const float* __restrict__ bias,       // [NH,N,K2]
                     float* __restrict__ out)              // [B,NH,N,D]
{
    __shared__ float tok_s[D_][KPAD];      // 16 KB, tokens[h] zero-padded
    __shared__ float attn_s[TILE_M][KPAD]; // 4 KB, attn tile (+bias +local); rows 256B-aligned

    const int tid  = threadIdx.x;
    const int lane = tid & 31;
    const int wave = tid >> 5;

    const int tilesPerBH = N_ / TILE_M;
    const int bh     = blockIdx.x / tilesPerBH;
    const int n_base = (blockIdx.x % tilesPerBH) * TILE_M;
    const int h      = bh % NH_;

    // ---- Stage tokens[h] -> LDS (zero-pad k to 64) -----------------------
    {
        const float* tk = tokens + (size_t)h * D_ * K2_;
        for (int i = tid; i < D_ * KPAD; i += 256) {
            int d = i >> 6;
            int k = i & 63;
            tok_s[d][k] = (k < K2_) ? tk[d * K2_ + k] : 0.0f;
        }
    }
    __syncthreads();

    if (wave < 4) {
        // ---- Waves 0..3: attn_tile[16][wave*16..+16] via FP32 WMMA -------
        const int   col_base = wave * 16;
        const float* qrow = q + (((size_t)bh * N_) + n_base) * D_;
        const int m  = lane & 15;   // M row (A) / N col (B,C,D)
        const int kp = lane >> 4;   // which K pair / which M half for C

        v8f acc = {};
        #pragma unroll
        for (int kk = 0; kk < D_; kk += 4) {
            v2f a, bm;
            // A 16x4 fragment: VGPR0 = K=kk+2*kp, VGPR1 = K=kk+2*kp+1
            a.x = qrow[(size_t)m * D_ + kk + 2 * kp + 0];
            a.y = qrow[(size_t)m * D_ + kk + 2 * kp + 1];
            // B 4x16 fragment (from LDS, already zero-padded past k2=49)
            bm.x = tok_s[kk + 2 * kp + 0][col_base + m];
            bm.y = tok_s[kk + 2 * kp + 1][col_base + m];
            acc = __builtin_amdgcn_wmma_f32_16x16x4_f32(
                      /*neg_a=*/false, a, /*neg_b=*/false, bm,
                      /*c_mod=*/(short)0, acc,
                      /*reuse_a=*/false, /*reuse_b=*/false);
        }

        // C/D layout: VGPR r -> M = r + 8*(lane>=16), N = lane%16
        #pragma unroll
        for (int r = 0; r < 8; ++r) {
            const int mm = r + 8 * kp;
            const int k  = col_base + m;
            if (k < K2_) {
                const int n = n_base + mm;
                float val = acc[r]
                          + bias[((size_t)h * N_ + n) * K2_ + k]
                          + attn_local[((size_t)bh * N_ + n) * K2_ + k];
                attn_s[mm][k] = val;
            }
        }
    } else {
        // ---- Waves 4..7: prefetch this block's v_local tile (200 KB) ----
        // Tile is contiguous: 16 pixels * 64 d * 49 k floats.
        const float* vtile = v_local + (((size_t)bh * N_ + n_base) * D_) * (size_t)K2_;
        const int total = TILE_M * D_ * K2_;          // 50176 floats
        for (int off = (tid - 128) * 32; off < total; off += 128 * 32) {
            __builtin_prefetch(vtile + off, 0, 0);    // global_prefetch_b8
        }
    }
    __syncthreads();

    // ---- AV aggregation: out[n,d] = sum_k attn_s[p][k] * v[n][d][k] ------
    {
        const int d  = tid & 63;
        const int p0 = tid >> 6;            // 0..3
        #pragma unroll
        for (int pi = 0; pi < 4; ++pi) {
            const int p = p0 * 4 + pi;      // covers 0..15; uniform row per wave
            const int n = n_base + p;
            const float* vrow = v_local +
                ((((size_t)bh * N_) + n) * D_ + d) * (size_t)K2_;

            // Register-cache the attention row with ds_load_b128's
            // (attn_s rows are 256B-aligned); the 12 quads cover k=0..47;
            // k=48 is read as a scalar.
            float4 av[12];
            const float4* arow4 = (const float4*)(&attn_s[p][0]);
            #pragma unroll
            for (int i = 0; i < 12; ++i) av[i] = arow4[i];
            const float a48 = attn_s[p][48];

            float s = 0.0f;
            #pragma unroll
            for (int i = 0; i < 12; ++i) {
                s = fmaf(av[i].x, vrow[4 * i + 0], s);
                s = fmaf(av[i].y, vrow[4 * i + 1], s);
                s = fmaf(av[i].z, vrow[4 * i + 2], s);
                s = fmaf(av[i].w, vrow[4 * i + 3], s);
            }
            s = fmaf(a48, vrow[48], s);

            out[(((size_t)bh) * N_ + n) * D_ + d] = s;
        }
    }
}

extern "C" void kernel_launch(void* const* d_in, const int* in_sizes, int n_in,
                              void* d_out, int out_size, void* d_ws, size_t ws_size,
                              hipStream_t stream) {
    const float* q          = (const float*)d_in[0];
    const float* attn_local = (const float*)d_in[1];
    const float* v_local    = (const float*)d_in[2];
    const float* tokens     = (const float*)d_in[3];
    const float* bias       = (const float*)d_in[4];
    float* out = (float*)d_out;

    const int n_blocks = B_ * NH_ * (N_ / TILE_M);   // 4096 workgroups
    swattn_av_fused<<<dim3(n_blocks), dim3(256), 0, stream>>>(
        q, attn_local, v_local, tokens, bias, out);
}